// DNM_76888504533025
// MI455X (gfx1250) — compile-verified
//
#include <hip/hip_runtime.h>
#include <math.h>

typedef __attribute__((ext_vector_type(2))) float v2f;
typedef __attribute__((ext_vector_type(8))) float v8f;

#define B_   64
#define O_   256
#define M_   8
#define D_   512
#define EPS_ 1e-5f
#define BT   16            // batch tile (WMMA M dim)
#define NBT  (B_ / BT)     // 4 batch tiles
#define NW   8             // waves per workgroup in main kernel
#define NJ   (D_ / (4 * NW))   // 16 uniform inner iterations per m-row

__device__ __forceinline__ float fast_sigmoid(float x) {
#if __has_builtin(__builtin_amdgcn_tanhf)
    // 1 TRANS op per element: sigmoid(x) = 0.5*tanh(x/2) + 0.5  (v_tanh_f32)
    return fmaf(0.5f, __builtin_amdgcn_tanhf(0.5f * x), 0.5f);
#else
    // 2 TRANS ops per element: v_exp_f32 + v_rcp_f32 (no IEEE div sequence)
    return __builtin_amdgcn_rcpf(1.0f + __expf(-x));
#endif
}

// ---------------------------------------------------------------------------
// Kernel 1: blocks 0..63  : xn = LayerNorm(x) over D (per row)
//           blocks 64..71 : wsum[m] = sum_d dn_w[m,:], bsum[m] = sum_d dn_b[m,:]
// ---------------------------------------------------------------------------
__global__ __launch_bounds__(256) void dnm_prep(
    const float* __restrict__ x,   const float* __restrict__ g,
    const float* __restrict__ be,  const float* __restrict__ dnw,
    const float* __restrict__ dnb, float* __restrict__ xn,
    float* __restrict__ wsum,      float* __restrict__ bsum)
{
    __shared__ float s1[256];
    __shared__ float s2[256];
    const int t  = threadIdx.x;
    const int bx = blockIdx.x;

    if (bx < B_) {
        const float* row = x + bx * D_;
        float a = row[t];
        float b = row[t + 256];
        s1[t] = a + b;
        s2[t] = a * a + b * b;
        __syncthreads();
        for (int s = 128; s > 0; s >>= 1) {
            if (t < s) { s1[t] += s1[t + s]; s2[t] += s2[t + s]; }
            __syncthreads();
        }
        float mu   = s1[0] * (1.0f / D_);
        float var  = s2[0] * (1.0f / D_) - mu * mu;
        float rstd = rsqrtf(var + EPS_);
        float* out = xn + bx * D_;
        out[t]       = (a - mu) * rstd * g[t]       + be[t];
        out[t + 256] = (b - mu) * rstd * g[t + 256] + be[t + 256];
    } else {
        const int m = bx - B_;
        const float* wr = dnw + m * D_;
        const float* br = dnb + m * D_;
        s1[t] = wr[t] + wr[t + 256];
        s2[t] = br[t] + br[t + 256];
        __syncthreads();
        for (int s = 128; s > 0; s >>= 1) {
            if (t < s) { s1[t] += s1[t + s]; s2[t] += s2[t + s]; }
            __syncthreads();
        }
        if (t == 0) { wsum[m] = s1[0]; bsum[m] = s2[0]; }
    }
}

// ---------------------------------------------------------------------------
// Kernel 2 (main): one workgroup per (o, batch-tile); 8 waves split the
// flattened (m,d) K axis (4096 positions, 4 per WMMA f32 step).
// Per step:  z = sigmoid(sw*xn + sb)   (VALU/TRANS, dominant cost)
//   c1 += Z x B   B col0 = ones (-> sum z), col m+1 = dn_w (-> T[m])
//   c2 += Z x Z^T (diag -> sum z^2)    -- reuses A registers as B operand
// Per-lane K offsets are folded into base pointers; the inner loop is a
// UNIFORM compile-time trip (16 iters, stride 32 floats) so it unrolls and
// addresses become base + immediate offsets (no per-iteration address ALU,
// no exec-mask loop management).
// ---------------------------------------------------------------------------
__global__ __launch_bounds__(NW * 32) void dnm_main(
    const float* __restrict__ xn,   const float* __restrict__ sw,
    const float* __restrict__ sb,   const float* __restrict__ dnw,
    const float* __restrict__ wsum, const float* __restrict__ bsum,
    float* __restrict__ pre)
{
    constexpr int XP = D_ + 4;                 // 516-float pitch: conflict-free b64 LDS reads
    __shared__ float xs[BT * XP];
    __shared__ float c1buf[NW * 256];
    __shared__ float c2buf[NW * 256];

    const int o     = blockIdx.x >> 2;         // / NBT
    const int b0    = (blockIdx.x & (NBT - 1)) * BT;
    const int tid   = threadIdx.x;
    const int w     = tid >> 5;
    const int lane  = tid & 31;
    const int tb    = lane & 15;               // batch row (A) / output col (B,D)
    const int khalf = lane >> 4;               // 0: K={0,1}, 1: K={2,3} (f32 A/B layout)
    const int kb    = khalf * 2;

    // Stage xn tile (16 x 512) into LDS, coalesced.
    for (int idx = tid; idx < BT * D_; idx += NW * 32) {
        int r = idx >> 9, c = idx & (D_ - 1);
        xs[r * XP + c] = xn[(b0 + r) * D_ + c];
    }
    __syncthreads();

    v8f c1 = {};
    v8f c2 = {};

    // Per-lane K start within a row, folded into base pointers once.
    const int    dstart = 4 * w + kb;
    const float  fone   = (tb == 0) ? 1.0f : 0.0f;
    const float* dwl    = dnw + (((tb >= 1) && (tb <= M_)) ? (tb - 1) * D_ : 0) + dstart;
    const float* xrow   = xs + tb * XP + dstart;
    const float* swl    = sw + (size_t)o * (M_ * D_) + dstart;
    const float* sbl    = sb + (size_t)o * (M_ * D_) + dstart;

    #pragma unroll 1
    for (int m = 0; m < M_; ++m) {
        __builtin_prefetch(swl + D_, 0, 1);    // global_prefetch_b8: next m row
        __builtin_prefetch(sbl + D_, 0, 1);
        const float fm = (tb == m + 1) ? 1.0f : 0.0f;
        #pragma unroll 4
        for (int j = 0; j < NJ; ++j) {         // uniform trip, stride 32 floats
            const int off = j * 4 * NW;
            float2 swv = *(const float2*)(swl + off);
            float2 sbv = *(const float2*)(sbl + off);
            float2 dwv = *(const float2*)(dwl + off);
            float2 xv  = *(const float2*)(xrow + off);  // ds_load_b64, conflict-free

            v2f a;
            a.x = fast_sigmoid(fmaf(swv.x, xv.x, sbv.x));
            a.y = fast_sigmoid(fmaf(swv.y, xv.y, sbv.y));

            v2f bm;
            bm.x = fmaf(fm, dwv.x, fone);
            bm.y = fmaf(fm, dwv.y, fone);

            c1 = __builtin_amdgcn_wmma_f32_16x16x4_f32(false, a, false, bm, (short)0, c1, false, false);
            c2 = __builtin_amdgcn_wmma_f32_16x16x4_f32(false, a, false, a,  (short)0, c2, false, false);
        }
        swl += D_;
        sbl += D_;
    }

    // Dump per-wave accumulators (C layout: VGPR r -> row r + khalf*8, col = tb).
    #pragma unroll
    for (int r = 0; r < 8; ++r) {
        const int row = r + khalf * 8;
        c1buf[w * 256 + row * 16 + tb] = c1[r];
        c2buf[w * 256 + row * 16 + tb] = c2[r];
    }
    __syncthreads();

    if (tid < BT) {
        float S1 = 0.f, S2 = 0.f, T[M_];
        #pragma unroll
        for (int m = 0; m < M_; ++m) T[m] = 0.f;
        for (int wv = 0; wv < NW; ++wv) {
            const float* p1 = c1buf + wv * 256 + tid * 16;
            const float* p2 = c2buf + wv * 256 + tid * 16;
            S1 += p1[0];
            S2 += p2[tid];                      // Gram diagonal
            #pragma unroll
            for (int m = 0; m < M_; ++m) T[m] += p1[m + 1];
        }
        const float inv  = 1.0f / (float)(M_ * D_);
        const float mu   = S1 * inv;
        const float var  = S2 * inv - mu * mu;
        const float rstd = rsqrtf(var + EPS_);
        float acc = 0.f;
        #pragma unroll
        for (int m = 0; m < M_; ++m) {
            float s = rstd * (T[m] - mu * wsum[m]) + bsum[m];
            acc += fast_sigmoid(s);
        }
        pre[(size_t)(b0 + tid) * O_ + o] = acc;
    }
}

// ---------------------------------------------------------------------------
// Kernel 3: softmax over o per batch row.
// ---------------------------------------------------------------------------
__global__ __launch_bounds__(256) void dnm_softmax(
    const float* __restrict__ pre, float* __restrict__ out)
{
    __shared__ float sh[256];
    const int b = blockIdx.x, t = threadIdx.x;
    float v = pre[b * O_ + t];
    sh[t] = v;
    __syncthreads();
    for (int s = 128; s > 0; s >>= 1) {
        if (t < s) sh[t] = fmaxf(sh[t], sh[t + s]);
        __syncthreads();
    }
    float mx = sh[0];
    __syncthreads();
    float e = __expf(v - mx);
    sh[t] = e;
    __syncthreads();
    for (int s = 128; s > 0; s >>= 1) {
        if (t < s) sh[t] += sh[t + s];
        __syncthreads();
    }
    out[b * O_ + t] = e * __builtin_amdgcn_rcpf(sh[0]);
}

// ---------------------------------------------------------------------------
extern "C" void kernel_launch(void* const* d_in, const int* in_sizes, int n_in,
                              void* d_out, int out_size, void* d_ws, size_t ws_size,
                              hipStream_t stream) {
    const float* x    = (const float*)d_in[0];  // (64, 512)
    const float* sn0w = (const float*)d_in[1];  // (512,)
    const float* sn0b = (const float*)d_in[2];  // (512,)
    const float* sw   = (const float*)d_in[3];  // (256, 8, 512)
    const float* sb   = (const float*)d_in[4];  // (256, 8, 512)
    const float* dnw  = (const float*)d_in[5];  // (8, 512)
    const float* dnb  = (const float*)d_in[6];  // (8, 512)
    float* out = (float*)d_out;                 // (64, 256)

    float* ws   = (float*)d_ws;
    float* xn   = ws;            // 64*512 = 32768 floats
    float* wsum = ws + 32768;    // 8
    float* bsum = ws + 32776;    // 8
    float* pre  = ws + 32784;    // 64*256 = 16384 floats

    dnm_prep   <<<B_ + M_,  256,     0, stream>>>(x, sn0w, sn0b, dnw, dnb, xn, wsum, bsum);
    dnm_main   <<<O_ * NBT, NW * 32, 0, stream>>>(xn, sw, sb, dnw, wsum, bsum, pre);
    dnm_softmax<<<B_,       256,     0, stream>>>(pre, out);
}